// vaup_43980465111582
// MI455X (gfx1250) — compile-verified
//
#include <hip/hip_runtime.h>

// ---------------------------------------------------------------------------
// out = X^T diag(w) X + va,   X: (BS=16384, D=512) fp32 row-major
//   w[b] = p*(1-p),  p = 1 + exp(-X[b,:] . mean)
//
// Pass 1: per-row weights into d_ws[0 .. BS)                     (64 KB)
// Pass 2: split-K (z=0..3) fp32-WMMA SYRK, double-buffered LDS,
//         partials into d_ws[BS + z*512*512 ...)                 (4 MB)
// Pass 3: out = sum_z partial[z] + va   (deterministic reduction)
// ---------------------------------------------------------------------------

#define DIM 512
#define BS  16384
#define KC  64                  // K-chunk depth staged through LDS
#define LDS_STRIDE 68           // 64 + 4 pad: 16B-aligned rows, bank spread
#define SPLITK 4
#define KSEG   (BS / SPLITK)    // 4096 K per z-slice
#define CHUNKS (KSEG / KC)      // 64 chunks per z-slice
#define CSIZE  (DIM * DIM)      // 262144 elements per partial

typedef __attribute__((ext_vector_type(2))) float v2f;
typedef __attribute__((ext_vector_type(8))) float v8f;

// ---------------- Pass 1: per-row weights --------------------------------
__global__ __launch_bounds__(256)
void vaup_weights_kernel(const float* __restrict__ xt,
                         const float* __restrict__ mean,
                         float* __restrict__ wts) {
    const int lane = threadIdx.x & 31;
    const int wave = threadIdx.x >> 5;
    const int row  = blockIdx.x * (blockDim.x >> 5) + wave;   // one wave32 per row
    const float* xr = xt + (long)row * DIM;

    float sum = 0.0f;
#pragma unroll
    for (int q = 0; q < 4; ++q) {            // 4 x float4 per lane = 512/wave
        const int base = q * 128 + lane * 4; // coalesced 512B chunks
        float4 x = *(const float4*)(xr + base);
        float4 m = *(const float4*)(mean + base);
        sum += x.x * m.x + x.y * m.y + x.z * m.z + x.w * m.w;
    }
#pragma unroll
    for (int off = 16; off > 0; off >>= 1)
        sum += __shfl_xor(sum, off, 32);

    if (lane == 0) {
        const float p = 1.0f + expf(-sum);   // faithful: NOT logistic sigmoid
        wts[row] = p * (1.0f - p);
    }
}

// ---------------- Pass 2: split-K weighted SYRK via fp32 WMMA ------------
// grid (8,8,SPLITK), 128 threads = 4 waves (2x2); wave -> 32x32 of C via
// 2x2 v8f accumulators. Double-buffered LDS chunks, one barrier per chunk.
__global__ __launch_bounds__(128)
void vaup_syrk_kernel(const float* __restrict__ xt,
                      const float* __restrict__ wts,
                      float* __restrict__ part) {
    __shared__ float As[2][KC][LDS_STRIDE];   // w[k] * X[k][i0..i0+63]
    __shared__ float Bs[2][KC][LDS_STRIDE];   //        X[k][j0..j0+63]

    const int tid  = threadIdx.x;
    const int lane = tid & 31;
    const int wave = tid >> 5;
    const int i0 = blockIdx.y * 64;
    const int j0 = blockIdx.x * 64;
    const int z  = blockIdx.z;
    const int wm = (wave >> 1) * 32;
    const int wn = (wave & 1) * 32;
    const int halfsel = lane >> 4;          // 0: lanes 0-15, 1: lanes 16-31
    const int lmod    = lane & 15;
    const int kbase = z * KSEG;
    const int kend  = kbase + KSEG;

    v8f acc[2][2];
#pragma unroll
    for (int a = 0; a < 2; ++a)
#pragma unroll
        for (int b = 0; b < 2; ++b)
#pragma unroll
            for (int r = 0; r < 8; ++r) acc[a][b][r] = 0.0f;

    // Stage one KCx64 A-tile (w-scaled) + B-tile into LDS buffer `bufi`.
    auto stage = [&](int bufi, int kb) {
#pragma unroll
        for (int r = 0; r < 8; ++r) {        // 1024 float4 per tile / 128 thr
            const int idx  = tid + 128 * r;  // 0..1023
            const int k    = idx >> 4;       // 0..63
            const int ccol = (idx & 15) << 2;
            const long gk  = (long)(kb + k) * DIM;
            const float w  = wts[kb + k];
            float4 xa = *(const float4*)(xt + gk + i0 + ccol);
            float4 xb = *(const float4*)(xt + gk + j0 + ccol);
            *(float4*)&As[bufi][k][ccol] =
                make_float4(w * xa.x, w * xa.y, w * xa.z, w * xa.w);
            *(float4*)&Bs[bufi][k][ccol] = xb;
            if (kb + 2 * KC < kend)          // warm L2 two chunks ahead
                __builtin_prefetch(xt + gk + (long)(2 * KC) * DIM + i0 + ccol, 0, 1);
        }
    };

    // Consume buffer `bufi`: 16 K-steps x 4 WMMAs per wave.
    auto compute = [&](int bufi) {
#pragma unroll 4
        for (int kk = 0; kk < KC; kk += 4) {
            const int krow = kk + (halfsel << 1);   // ISA A/B lane K-pair
            v2f a[2], b[2];
#pragma unroll
            for (int t = 0; t < 2; ++t) {
                const int il = wm + t * 16 + lmod;
                a[t][0] = As[bufi][krow][il];
                a[t][1] = As[bufi][krow + 1][il];
                const int jl = wn + t * 16 + lmod;
                b[t][0] = Bs[bufi][krow][jl];
                b[t][1] = Bs[bufi][krow + 1][jl];
            }
#pragma unroll
            for (int tm = 0; tm < 2; ++tm)
#pragma unroll
                for (int tn = 0; tn < 2; ++tn)
                    acc[tm][tn] = __builtin_amdgcn_wmma_f32_16x16x4_f32(
                        false, a[tm], false, b[tn],
                        (short)0, acc[tm][tn], false, false);
        }
    };

    stage(0, kbase);
    int buf = 0;
    for (int c = 0; c < CHUNKS; ++c) {
        __syncthreads();                     // staged buf visible; prev reads done
        if (c + 1 < CHUNKS) stage(buf ^ 1, kbase + (c + 1) * KC);
        compute(buf);
        buf ^= 1;
    }

    // Epilogue: C layout -> VGPR r: lanes0-15 M=r, lanes16-31 M=r+8; N=lane%16.
    float* po = part + (long)z * CSIZE;
#pragma unroll
    for (int tm = 0; tm < 2; ++tm) {
#pragma unroll
        for (int tn = 0; tn < 2; ++tn) {
            const int Mbase = i0 + wm + tm * 16 + (halfsel << 3);
            const int N     = j0 + wn + tn * 16 + lmod;
#pragma unroll
            for (int r = 0; r < 8; ++r)
                po[(long)(Mbase + r) * DIM + N] = acc[tm][tn][r];
        }
    }
}

// ---------------- Pass 3: deterministic split-K reduction + va -----------
__global__ __launch_bounds__(256)
void vaup_reduce_kernel(const float* __restrict__ part,
                        const float* __restrict__ va,
                        float* __restrict__ out) {
    const int idx = (blockIdx.x * 256 + threadIdx.x) * 4;
    float4 s = *(const float4*)(va + idx);
#pragma unroll
    for (int zz = 0; zz < SPLITK; ++zz) {
        float4 p = *(const float4*)(part + (long)zz * CSIZE + idx);
        s.x += p.x; s.y += p.y; s.z += p.z; s.w += p.w;
    }
    *(float4*)(out + idx) = s;
}

// ---------------------------------------------------------------------------
extern "C" void kernel_launch(void* const* d_in, const int* in_sizes, int n_in,
                              void* d_out, int out_size, void* d_ws, size_t ws_size,
                              hipStream_t stream) {
    const float* mean = (const float*)d_in[0];   // (512,)
    const float* va   = (const float*)d_in[1];   // (512,512)
    const float* xt   = (const float*)d_in[2];   // (16384,512)
    float* out  = (float*)d_out;                 // (512,512)
    float* wts  = (float*)d_ws;                  // [0, BS) floats
    float* part = wts + BS;                      // SPLITK x 512x512 partials

    vaup_weights_kernel<<<BS / 8, 256, 0, stream>>>(xt, mean, wts);

    dim3 grid(DIM / 64, DIM / 64, SPLITK);       // 256 workgroups
    vaup_syrk_kernel<<<grid, 128, 0, stream>>>(xt, wts, part);

    vaup_reduce_kernel<<<CSIZE / (4 * 256), 256, 0, stream>>>(part, va, out);
}